// DWRModel_40037685133330
// MI455X (gfx1250) — compile-verified
//
#include <hip/hip_runtime.h>
#include <math.h>

// ---------------------------------------------------------------------------
// DWR graph model pipeline for MI455X (gfx1250, wave32).
// Dense GEMMs use V_WMMA_F32_16X16X4_F32 (exact f32 matrix math).
// Edge stages are single-pass with L2-resident float atomics.
// ---------------------------------------------------------------------------

typedef __attribute__((ext_vector_type(2))) float v2f;
typedef __attribute__((ext_vector_type(8))) float v8f;

// Monotone float <-> orderable-uint encoding for atomicMax on floats.
__device__ __forceinline__ unsigned ordf(float f) {
  unsigned u = __float_as_uint(f);
  return (u & 0x80000000u) ? ~u : (u | 0x80000000u);
}
__device__ __forceinline__ float unordf(unsigned u) {
  return (u & 0x80000000u) ? __uint_as_float(u & 0x7fffffffu)
                           : __uint_as_float(~u);
}

// ---------------- init ----------------
__global__ void k_init_nodes(float* __restrict__ deno, float* __restrict__ deg,
                             float* __restrict__ z, unsigned* __restrict__ attmax,
                             int n) {
  int i = blockIdx.x * blockDim.x + threadIdx.x;
  if (i < n) { deno[i] = 0.0f; deg[i] = 1.0f; z[i] = 0.0f; }
  if (i == 0) *attmax = 0u;  // encoded minimum (very negative float)
}

__global__ void k_zero(float* __restrict__ p, long long n) {
  long long i = (long long)blockIdx.x * blockDim.x + threadIdx.x;
  if (i < n) p[i] = 0.0f;
}

// ---------------- WMMA GEMM: out[nrows x 32] = act(A[nrows x KDIM] @ W[KDIM x 32] + bias)
// One wave = one 16-row tile, full 32 output columns (two 16x16 accumulators).
template <int KDIM, bool RELU, bool BIAS>
__launch_bounds__(256)
__global__ void k_gemm32(const float* __restrict__ A, const float* __restrict__ W,
                         const float* __restrict__ bias, float* __restrict__ out,
                         int nrows) {
  const int lane = threadIdx.x & 31;
  const int wave = threadIdx.x >> 5;
  const int tile = blockIdx.x * (blockDim.x >> 5) + wave;
  const int m0 = tile * 16;
  if (m0 >= nrows) return;                 // wave-uniform: EXEC stays all-ones for WMMA
  const int mr   = lane & 15;              // A row within tile / B,C column
  const int half = lane >> 4;              // lane half selects K/M grouping

  v8f c0 = {}; v8f c1 = {};
  const int arow_i = min(m0 + mr, nrows - 1);
  const float* arow = A + (long long)arow_i * KDIM + 2 * half;

#pragma unroll
  for (int k0 = 0; k0 < KDIM; k0 += 4) {
    // A fragment: lane holds A[m][k0 + 2*half + {0,1}]
    v2f a; a.x = arow[k0]; a.y = arow[k0 + 1];
    // B fragments: lane holds W[k0 + 2*half + v][n] for n = mr (tile0) / 16+mr (tile1)
    const float* wb = W + (k0 + 2 * half) * 32 + mr;
    v2f b0, b1;
    b0.x = wb[0];  b0.y = wb[32];
    b1.x = wb[16]; b1.y = wb[48];
    c0 = __builtin_amdgcn_wmma_f32_16x16x4_f32(false, a, false, b0, (short)0, c0, false, false);
    c1 = __builtin_amdgcn_wmma_f32_16x16x4_f32(false, a, false, b1, (short)0, c1, false, false);
  }

  const float bb0 = BIAS ? bias[mr]      : 0.0f;
  const float bb1 = BIAS ? bias[16 + mr] : 0.0f;
  float* orow = out + (long long)m0 * 32;
#pragma unroll
  for (int v = 0; v < 8; ++v) {
    const int m = v + 8 * half;            // C/D layout: row = v + 8*half, col = mr
    if (m0 + m < nrows) {
      float r0 = c0[v] + bb0;
      float r1 = c1[v] + bb1;
      if (RELU) { r0 = fmaxf(r0, 0.0f); r1 = fmaxf(r1, 0.0f); }
      orow[(long long)m * 32 + mr]      = r0;
      orow[(long long)m * 32 + 16 + mr] = r1;
    }
  }
}

// ---------------- edge pass 1: raw attention dot, degree count, global max
__global__ void k_att(const float* __restrict__ h, const int* __restrict__ row,
                      const int* __restrict__ col, float* __restrict__ att,
                      float* __restrict__ deg, unsigned* __restrict__ attmax, int E) {
  const int e = blockIdx.x * blockDim.x + threadIdx.x;
  float dot = -3.402823466e38f;
  if (e < E) {
    __builtin_prefetch(row + e + 16384, 0, 0);   // global_prefetch_b8 on the streamed lists
    __builtin_prefetch(col + e + 16384, 0, 0);
    const int r = row[e], c = col[e];
    const float4* hr = (const float4*)(h + (long long)r * 32);
    const float4* hc = (const float4*)(h + (long long)c * 32);
    float acc = 0.0f;
#pragma unroll
    for (int q = 0; q < 8; ++q) {
      const float4 a = hr[q], b = hc[q];
      acc += a.x * b.x + a.y * b.y + a.z * b.z + a.w * b.w;
    }
    att[e] = acc;
    dot = acc;
    atomicAdd(&deg[c], 1.0f);                    // deg = segment_sum(1, col) (+1 via init)
  }
  // wave32 max reduce, then one atomic per wave
#pragma unroll
  for (int off = 16; off > 0; off >>= 1)
    dot = fmaxf(dot, __shfl_xor(dot, off, 32));
  if ((threadIdx.x & 31) == 0) atomicMax(attmax, ordf(dot));
}

// ---------------- edge pass 2: exp(att - max), softmax denominator
__global__ void k_expdeno(const int* __restrict__ row, float* __restrict__ att,
                          const unsigned* __restrict__ attmax,
                          float* __restrict__ deno, int E) {
  const int e = blockIdx.x * blockDim.x + threadIdx.x;
  if (e >= E) return;
  __builtin_prefetch(row + e + 16384, 0, 0);
  const float mx = unordf(*attmax);
  const float a = __expf(att[e] - mx);
  att[e] = a;
  atomicAdd(&deno[row[e]], a);
}

// ---------------- edge pass 3: exposure z = segsum(att_norm * t[col], row)
__global__ void k_z(const int* __restrict__ row, const int* __restrict__ col,
                    const float* __restrict__ att, const float* __restrict__ deno,
                    const float* __restrict__ t, float* __restrict__ z, int E) {
  const int e = blockIdx.x * blockDim.x + threadIdx.x;
  if (e >= E) return;
  __builtin_prefetch(row + e + 16384, 0, 0);
  __builtin_prefetch(col + e + 16384, 0, 0);
  const int r = row[e];
  const float a = att[e] / (deno[r] + 1e-8f);
  atomicAdd(&z[r], a * t[col[e]]);
}

// ---------------- edge pass 4: agg[col] += dis[row] * hw[row]  (one wave = one edge)
__global__ void k_agg(const int* __restrict__ row, const int* __restrict__ col,
                      const float* __restrict__ hw, const float* __restrict__ deg,
                      float* __restrict__ agg, long long EF) {
  const long long i = (long long)blockIdx.x * blockDim.x + threadIdx.x;
  if (i >= EF) return;
  const int e = (int)(i >> 5);
  const int f = (int)(i & 31);
  const int r = row[e], c = col[e];
  const float dis = rsqrtf(deg[r]);
  atomicAdd(&agg[(long long)c * 32 + f], dis * hw[(long long)r * 32 + f]);
}

// ---------------- node residual: rep = h + relu(dis*agg + dis^2*hw + b_gcn)
__global__ void k_rep(const float* __restrict__ h, const float* __restrict__ hw,
                      const float* __restrict__ agg, const float* __restrict__ deg,
                      const float* __restrict__ b_gcn, float* __restrict__ rep,
                      long long NF) {
  const long long i = (long long)blockIdx.x * blockDim.x + threadIdx.x;
  if (i >= NF) return;
  const int node = (int)(i >> 5);
  const int f = (int)(i & 31);
  const float dis = rsqrtf(deg[node]);
  const float g = dis * agg[i] + dis * dis * hw[i] + b_gcn[f];
  rep[i] = h[i] + fmaxf(g, 0.0f);
}

// ---------------- MLP head: out = relu([rep,t,z] @ W1 + b1) @ W2 + b2
__global__ void k_head(const float* __restrict__ rep, const float* __restrict__ t,
                       const float* __restrict__ z, const float* __restrict__ W1,
                       const float* __restrict__ b1, const float* __restrict__ W2,
                       const float* __restrict__ b2, float* __restrict__ out, int n) {
  const int i = blockIdx.x * blockDim.x + threadIdx.x;
  if (i >= n) return;
  float r[32];
  const float4* rp = (const float4*)(rep + (long long)i * 32);
#pragma unroll
  for (int q = 0; q < 8; ++q) {
    const float4 v = rp[q];
    r[4 * q + 0] = v.x; r[4 * q + 1] = v.y; r[4 * q + 2] = v.z; r[4 * q + 3] = v.w;
  }
  const float ti = t[i], zi = z[i];
  float o = b2[0];
#pragma unroll 4
  for (int f = 0; f < 32; ++f) {
    float acc = b1[f] + ti * W1[32 * 32 + f] + zi * W1[33 * 32 + f];
#pragma unroll
    for (int k = 0; k < 32; ++k) acc += r[k] * W1[k * 32 + f];
    o += fmaxf(acc, 0.0f) * W2[f];
  }
  out[i] = o;
}

// ---------------------------------------------------------------------------
extern "C" void kernel_launch(void* const* d_in, const int* in_sizes, int n_in,
                              void* d_out, int out_size, void* d_ws, size_t ws_size,
                              hipStream_t stream) {
  const float* x     = (const float*)d_in[0];
  const float* t     = (const float*)d_in[1];
  const int*   row   = (const int*)d_in[2];
  const int*   col   = (const int*)d_in[3];
  const float* W_emb = (const float*)d_in[4];
  const float* b_emb = (const float*)d_in[5];
  const float* W_gcn = (const float*)d_in[6];
  const float* b_gcn = (const float*)d_in[7];
  const float* W1    = (const float*)d_in[8];
  const float* b1    = (const float*)d_in[9];
  const float* W2    = (const float*)d_in[10];
  const float* b2    = (const float*)d_in[11];

  const int N = in_sizes[1];   // t has N elements
  const int E = in_sizes[2];   // row has E elements

  // outputs: (out[N,1], rep[N,32], z[N]) concatenated flat
  float* out = (float*)d_out;
  float* rep = out + N;
  float* z   = rep + (long long)N * 32;

  // workspace layout (floats)
  float* ws   = (float*)d_ws;
  float* h    = ws;                          // N*32
  float* hw   = h   + (long long)N * 32;     // N*32
  float* agg  = hw  + (long long)N * 32;     // N*32
  float* deno = agg + (long long)N * 32;     // N
  float* deg  = deno + N;                    // N
  float* att  = deg  + N;                    // E
  unsigned* attmax = (unsigned*)(att + E);   // 1

  const int B = 256;
  const long long NF = (long long)N * 32;
  const long long EF = (long long)E * 32;
  const int tiles = (N + 15) / 16;
  const int gemm_blocks = (tiles + 7) / 8;   // 8 waves per block

  // init
  k_init_nodes<<<(N + B - 1) / B, B, 0, stream>>>(deno, deg, z, attmax, N);
  k_zero<<<(int)((NF + B - 1) / B), B, 0, stream>>>(agg, NF);

  // h = relu(x @ W_emb + b_emb)   [WMMA f32 16x16x4, K=128]
  k_gemm32<128, true, true><<<gemm_blocks, B, 0, stream>>>(x, W_emb, b_emb, h, N);

  // edge attention pipeline
  k_att<<<(E + B - 1) / B, B, 0, stream>>>(h, row, col, att, deg, attmax, E);
  k_expdeno<<<(E + B - 1) / B, B, 0, stream>>>(row, att, attmax, deno, E);
  k_z<<<(E + B - 1) / B, B, 0, stream>>>(row, col, att, deno, t, z, E);

  // hw = h @ W_gcn                [WMMA f32 16x16x4, K=32]
  k_gemm32<32, false, false><<<gemm_blocks, B, 0, stream>>>(h, W_gcn, nullptr, hw, N);

  // normalized scatter-aggregate
  k_agg<<<(int)((EF + B - 1) / B), B, 0, stream>>>(row, col, hw, deg, agg, EF);

  // rep = h + relu(gcn)
  k_rep<<<(int)((NF + B - 1) / B), B, 0, stream>>>(h, hw, agg, deg, b_gcn, rep, NF);

  // MLP head -> out
  k_head<<<(N + B - 1) / B, B, 0, stream>>>(rep, t, z, W1, b1, W2, b2, out, N);
}